// LocalAttnDecoderRNN_28071906246908
// MI455X (gfx1250) — compile-verified
//
#include <hip/hip_runtime.h>
#include <math.h>

typedef __attribute__((ext_vector_type(2))) float v2f;
typedef __attribute__((ext_vector_type(8))) float v8f;

#define H    1024
#define VSZ  128000
#define LMAX 4096

// ws layout (float offsets)
#define WS_EMB    0        // 1024
#define WS_ATTNW  1024     // 4096
#define WS_CENTER 5120     // 1
#define WS_NORM   5124     // 1
#define WS_AA     5632     // 1024
#define WS_X      6656     // 1024
#define WS_GATES  7680     // 4096
#define WS_RESID  11776    // 1024
#define WS_PM     12800    // 250
#define WS_PS     13056    // 250

// ---------------------------------------------------------------- embedding gather
__global__ void k_prep(const int* __restrict__ tok, const float* __restrict__ emb,
                       float* __restrict__ dst) {
  int t = blockIdx.x * blockDim.x + threadIdx.x;
  if (t < H) dst[t] = emb[(size_t)tok[0] * H + t];
}

// ---------------------------------------------------------------- WMMA f32 GEMV
// y[n] = b0[n] + b1[n] + sum_k W0[n*ldw0+k]*x0[k] + sum_k W1[n*ldw1+k]*x1[k]
// 256 threads = 8 waves, 16 rows/wave, 128 rows/block.
// NT: use non-temporal loads for W (huge single-use streams, keep L2 for
// the resident small weights).
template <bool NT>
__global__ void k_wmma_gemv(const float* __restrict__ W0, int ldw0,
                            const float* __restrict__ x0, int K0,
                            const float* __restrict__ W1, int ldw1,
                            const float* __restrict__ x1, int K1,
                            const float* __restrict__ b0, const float* __restrict__ b1,
                            float* __restrict__ y, int relu) {
  __shared__ __align__(16) float sx[3072];
  int tid = threadIdx.x;
  for (int i = tid; i < K0; i += 256) sx[i] = x0[i];
  if (W1) for (int i = tid; i < K1; i += 256) sx[K0 + i] = x1[i];
  __syncthreads();

  int wave = tid >> 5;
  int lane = tid & 31;
  int r    = lane & 15;     // row within 16-row tile
  int hi   = lane >> 4;     // K-pair select (0 -> K 0,1 ; 1 -> K 2,3)
  int co   = 2 * hi;
  int rowbase = (blockIdx.x * 8 + wave) * 16;

  v8f c = {0.f, 0.f, 0.f, 0.f, 0.f, 0.f, 0.f, 0.f};

  const float* Wr = W0 + (size_t)(rowbase + r) * (size_t)ldw0;
#pragma unroll 8
  for (int k = 0; k < K0; k += 4) {
    v2f a = NT ? __builtin_nontemporal_load((const v2f*)(Wr + k + co))
               : *(const v2f*)(Wr + k + co);
    v2f b = *(const v2f*)(sx + k + co);
    c = __builtin_amdgcn_wmma_f32_16x16x4_f32(false, a, false, b, (short)0, c, false, false);
  }
  if (W1) {
    const float* Wr1 = W1 + (size_t)(rowbase + r) * (size_t)ldw1;
    const float* sx1 = sx + K0;
#pragma unroll 8
    for (int k = 0; k < K1; k += 4) {
      v2f a = NT ? __builtin_nontemporal_load((const v2f*)(Wr1 + k + co))
                 : *(const v2f*)(Wr1 + k + co);
      v2f b = *(const v2f*)(sx1 + k + co);
      c = __builtin_amdgcn_wmma_f32_16x16x4_f32(false, a, false, b, (short)0, c, false, false);
    }
  }

  // D columns are all identical; lane 0 holds rows 0..7 (VGPR j = D[j][0]),
  // lane 16 holds rows 8..15.
  if (r == 0) {
    int roff = rowbase + hi * 8;
#pragma unroll
    for (int j = 0; j < 8; ++j) {
      float v = c[j];
      if (b0) v += b0[roff + j];
      if (b1) v += b1[roff + j];
      if (relu) v = fmaxf(v, 0.f);
      y[roff + j] = v;
    }
  }
}

// ---------------------------------------------------------------- attention center
__global__ void k_center(const float* __restrict__ aw, const float* __restrict__ alw,
                         const float* __restrict__ alb, float* __restrict__ outc) {
  __shared__ float red[1024];
  int t = threadIdx.x;
  float s = 0.f;
  for (int i = t; i < LMAX; i += 1024) s += aw[i] * alw[i];
  red[t] = s; __syncthreads();
  for (int off = 512; off > 0; off >>= 1) {
    if (t < off) red[t] += red[t + off];
    __syncthreads();
  }
  if (t == 0) outc[0] = red[0] + alb[0];
}

// ---------------------------------------------------------------- windowed attn_applied
// wv is nonzero only for |pos - center| < 5 -> touch <= 11 rows of E.
__global__ void k_attn_apply(const float* __restrict__ aw, const float* __restrict__ cen,
                             const float* __restrict__ E, float* __restrict__ aa) {
  float center = cen[0];
  int t = threadIdx.x;  // 256 threads, 4 cols each
  float4 acc = make_float4(0.f, 0.f, 0.f, 0.f);
  int ilo = (int)ceilf(center - 5.0f);  if (ilo < 0) ilo = 0;
  int ihi = (int)floorf(center + 5.0f); if (ihi > LMAX - 1) ihi = LMAX - 1;
  for (int i = ilo; i <= ihi; ++i) {
    float p  = (float)i;
    float wv = fmaxf(p - center + 5.f, 0.f) * fmaxf(center - p + 5.f, 0.f) * (1.f / 25.f);
    float la = aw[i] * wv;
    float4 e = *(const float4*)(E + (size_t)i * H + 4 * t);
    acc.x += la * e.x; acc.y += la * e.y; acc.z += la * e.z; acc.w += la * e.w;
  }
  *(float4*)(aa + 4 * t) = acc;
}

// ---------------------------------------------------------------- LSTM cell + residual
__global__ void k_lstm(const float* __restrict__ g, const float* __restrict__ c0,
                       const float* __restrict__ x, float* __restrict__ dout,
                       float* __restrict__ resid) {
  int j = blockIdx.x * blockDim.x + threadIdx.x;
  if (j >= H) return;
  float si = 1.f / (1.f + expf(-g[j]));
  float sf = 1.f / (1.f + expf(-g[H + j]));
  float tg = tanhf(g[2 * H + j]);
  float so = 1.f / (1.f + expf(-g[3 * H + j]));
  float cn = sf * c0[j] + si * tg;
  float hn = so * tanhf(cn);
  dout[VSZ + j]     = hn;   // h_new
  dout[VSZ + H + j] = cn;   // c_new
  resid[j] = hn + x[j];
}

// ---------------------------------------------------------------- log-softmax (online)
__global__ void k_sm_part(const float* __restrict__ lg, float* __restrict__ pm,
                          float* __restrict__ ps) {
  __shared__ float red[512];
  int t = threadIdx.x;
  float v = lg[blockIdx.x * 512 + t];
  red[t] = v; __syncthreads();
  for (int off = 256; off > 0; off >>= 1) {
    if (t < off) red[t] = fmaxf(red[t], red[t + off]);
    __syncthreads();
  }
  float m = red[0]; __syncthreads();
  red[t] = expf(v - m); __syncthreads();
  for (int off = 256; off > 0; off >>= 1) {
    if (t < off) red[t] += red[t + off];
    __syncthreads();
  }
  if (t == 0) { pm[blockIdx.x] = m; ps[blockIdx.x] = red[0]; }
}

__global__ void k_sm_comb(const float* __restrict__ pm, const float* __restrict__ ps,
                          float* __restrict__ norm, int nb) {
  __shared__ float red[256];
  int t = threadIdx.x;
  float m = (t < nb) ? pm[t] : -INFINITY;
  red[t] = m; __syncthreads();
  for (int off = 128; off > 0; off >>= 1) {
    if (t < off) red[t] = fmaxf(red[t], red[t + off]);
    __syncthreads();
  }
  float M = red[0]; __syncthreads();
  red[t] = (t < nb) ? ps[t] * expf(pm[t] - M) : 0.f;
  __syncthreads();
  for (int off = 128; off > 0; off >>= 1) {
    if (t < off) red[t] += red[t + off];
    __syncthreads();
  }
  if (t == 0) norm[0] = M + logf(red[0]);
}

__global__ void k_final(float* __restrict__ lg, const float* __restrict__ norm) {
  int i = blockIdx.x * blockDim.x + threadIdx.x;
  lg[i] -= norm[0];
}

// ---------------------------------------------------------------- launch
extern "C" void kernel_launch(void* const* d_in, const int* in_sizes, int n_in,
                              void* d_out, int out_size, void* d_ws, size_t ws_size,
                              hipStream_t stream) {
  const int*   tok   = (const int*)  d_in[0];
  const float* h0    = (const float*)d_in[1];
  const float* c0    = (const float*)d_in[2];
  const float* E     = (const float*)d_in[3];
  const float* emb   = (const float*)d_in[4];
  const float* attnW = (const float*)d_in[5];
  const float* attnB = (const float*)d_in[6];
  const float* alW   = (const float*)d_in[7];
  const float* alB   = (const float*)d_in[8];
  const float* combW = (const float*)d_in[9];
  const float* combB = (const float*)d_in[10];
  const float* Wih   = (const float*)d_in[11];
  const float* Whh   = (const float*)d_in[12];
  const float* bih   = (const float*)d_in[13];
  const float* bhh   = (const float*)d_in[14];
  const float* outW  = (const float*)d_in[15];
  const float* outB  = (const float*)d_in[16];
  float* out = (float*)d_out;
  float* ws  = (float*)d_ws;

  k_prep<<<4, 256, 0, stream>>>(tok, emb, ws + WS_EMB);

  // attn_weights = [emb, h] @ attn_W.T + attn_b      (N=4096, K=1024+1024)
  k_wmma_gemv<false><<<32, 256, 0, stream>>>(attnW, 2048, ws + WS_EMB, 1024,
                                             attnW + 1024, 2048, h0, 1024,
                                             attnB, nullptr, ws + WS_ATTNW, 0);

  k_center<<<1, 1024, 0, stream>>>(ws + WS_ATTNW, alW, alB, ws + WS_CENTER);
  k_attn_apply<<<1, 256, 0, stream>>>(ws + WS_ATTNW, ws + WS_CENTER, E, ws + WS_AA);

  // output = relu([emb, attn_applied] @ comb_W.T + comb_b)   (N=1024)
  k_wmma_gemv<false><<<8, 256, 0, stream>>>(combW, 2048, ws + WS_EMB, 1024,
                                            combW + 1024, 2048, ws + WS_AA, 1024,
                                            combB, nullptr, ws + WS_X, 1);

  // gates = x @ W_ih.T + h @ W_hh.T + b_ih + b_hh            (N=4096)
  k_wmma_gemv<false><<<32, 256, 0, stream>>>(Wih, 1024, ws + WS_X, 1024,
                                             Whh, 1024, h0, 1024,
                                             bih, bhh, ws + WS_GATES, 0);

  k_lstm<<<4, 256, 0, stream>>>(ws + WS_GATES, c0, ws + WS_X, out, ws + WS_RESID);

  // logits = resid @ out_W.T + out_b   (N=128000, 524 MB stream -> NT loads)
  k_wmma_gemv<true><<<1000, 256, 0, stream>>>(outW, 1024, ws + WS_RESID, 1024,
                                              nullptr, 1024, nullptr, 0,
                                              outB, nullptr, out, 0);

  k_sm_part<<<250, 512, 0, stream>>>(out, ws + WS_PM, ws + WS_PS);
  k_sm_comb<<<1, 256, 0, stream>>>(ws + WS_PM, ws + WS_PS, ws + WS_NORM, 250);
  k_final<<<500, 256, 0, stream>>>(out, ws + WS_NORM);
}